// SimilarityScorer_33457795236058
// MI455X (gfx1250) — compile-verified
//
#include <hip/hip_runtime.h>
#include <hip/hip_bf16.h>

typedef __attribute__((ext_vector_type(2))) float v2f;
typedef __attribute__((ext_vector_type(8))) float v8f;

#define D_FEAT 48

// Pass 1: per-node L2 norm + zero the scatter accumulators (ws is poisoned each run).
__global__ __launch_bounds__(256) void simscore_node_norms(
    const float* __restrict__ feat, float* __restrict__ norms,
    float* __restrict__ sums, float* __restrict__ cnt, int n_nodes) {
  int n = blockIdx.x * blockDim.x + threadIdx.x;
  if (n >= n_nodes) return;
  const float4* row = (const float4*)(feat + (size_t)n * D_FEAT);
  float acc = 0.f;
#pragma unroll
  for (int i = 0; i < D_FEAT / 4; ++i) {
    float4 v = row[i];
    acc += v.x * v.x + v.y * v.y + v.z * v.z + v.w * v.w;
  }
  norms[n] = sqrtf(acc);
  sums[n] = 0.f;
  cnt[n] = 0.f;
}

// Pass 2: 16 edges per wave. Dot products via V_WMMA_F32_16X16X4_F32:
//   A[m][k] = feat[src[tile*16+m]][kbase+k], B[k][n] = feat[dst[tile*16+n]][kbase+k]
// diag(C) after 12 accumulating WMMAs = the 16 per-edge dot products (f32 exact).
__global__ __launch_bounds__(256) void simscore_edge_wmma(
    const float* __restrict__ feat, const int* __restrict__ src,
    const int* __restrict__ dst, const float* __restrict__ norms,
    float* __restrict__ sums, float* __restrict__ cnt, int n_edges) {
  const int lane  = threadIdx.x & 31;
  const int wid   = (int)((blockIdx.x * blockDim.x + threadIdx.x) >> 5);
  const int nwav  = (int)((gridDim.x * blockDim.x) >> 5);
  const int slot  = lane & 15;        // which edge of the 16-edge tile this lane serves
  const int kb    = (lane >> 4) * 2;  // K sub-offset within the 16x4 A / 4x16 B layout
  const int ntiles = (n_edges + 15) >> 4;

  for (int t = wid; t < ntiles; t += nwav) {   // wave-uniform loop: EXEC all-1s at WMMA
    int e = t * 16 + slot;
    bool evalid = (e < n_edges);
    int ec = evalid ? e : 0;
    int si = src[ec];
    int di = dst[ec];
    const float* srow = feat + (size_t)si * D_FEAT + kb;
    const float* drow = feat + (size_t)di * D_FEAT + kb;

    v8f c = {};
#pragma unroll
    for (int i = 0; i < D_FEAT / 4; ++i) {
      v2f a = *(const v2f*)(srow + 4 * i);   // global_load_b64 gather (L2-resident table)
      v2f b = *(const v2f*)(drow + 4 * i);
      c = __builtin_amdgcn_wmma_f32_16x16x4_f32(
          /*neg_a=*/false, a, /*neg_b=*/false, b,
          /*c_mod=*/(short)0, c, /*reuse_a=*/false, /*reuse_b=*/false);
    }

    // Diagonal of the 16x16 f32 C: element (r,r) -> VGPR r, lane r;
    // element (8+r,8+r) -> VGPR r, lane 24+r.
    bool dvalid = (lane < 8) || (lane >= 24);
    int q = (lane < 8) ? lane : (lane - 24);
    float sim = c[0];
#pragma unroll
    for (int j = 1; j < 8; ++j) sim = (q == j) ? c[j] : sim;

    if (dvalid && evalid) {   // lanes holding a diagonal already loaded si/di for that slot
      float nn = norms[si] * norms[di];
      sim = sim / fmaxf(nn, 1e-8f);
      atomicAdd(&sums[di], sim);
      atomicAdd(&cnt[di], 1.0f);
    }
  }
}

// Pass 3: mean over in-edges, 0 for isolated nodes.
__global__ __launch_bounds__(256) void simscore_finalize(
    const float* __restrict__ sums, const float* __restrict__ cnt,
    float* __restrict__ out, int n_nodes) {
  int n = blockIdx.x * blockDim.x + threadIdx.x;
  if (n >= n_nodes) return;
  float k = cnt[n];
  out[n] = (k > 0.f) ? sums[n] / fmaxf(k, 1.0f) : 0.f;
}

extern "C" void kernel_launch(void* const* d_in, const int* in_sizes, int n_in,
                              void* d_out, int out_size, void* d_ws, size_t ws_size,
                              hipStream_t stream) {
  const float* feat = (const float*)d_in[0];
  const int*   src  = (const int*)d_in[1];
  const int*   dst  = (const int*)d_in[2];
  float*       out  = (float*)d_out;

  const int n_nodes = in_sizes[0] / D_FEAT;
  const int n_edges = in_sizes[1];

  float* norms = (float*)d_ws;        // [n_nodes]
  float* sums  = norms + n_nodes;     // [n_nodes]
  float* cnt   = sums + n_nodes;      // [n_nodes]

  const int B = 256;
  const int ngrid = (n_nodes + B - 1) / B;
  simscore_node_norms<<<ngrid, B, 0, stream>>>(feat, norms, sums, cnt, n_nodes);

  const int ntiles = (n_edges + 15) / 16;        // one 16-edge tile per wave
  int nblocks = (ntiles + 7) / 8;                // 8 waves per 256-thread block
  if (nblocks > 16384) nblocks = 16384;          // grid-stride covers the rest
  simscore_edge_wmma<<<nblocks, B, 0, stream>>>(feat, src, dst, norms, sums, cnt, n_edges);

  simscore_finalize<<<ngrid, B, 0, stream>>>(sums, cnt, out, n_nodes);
}